// GIBLiLayer_4337916970039
// MI455X (gfx1250) — compile-verified
//
#include <hip/hip_runtime.h>

typedef __attribute__((ext_vector_type(2))) float v2f;
typedef __attribute__((ext_vector_type(8))) float v8f;

#define NPTS   8192
#define NBATCH 4
#define OBS    16
#define KSIZE  0.2f
#define WAVES_PER_BLOCK 8
#define TILE_STRIDE 17   // 16 + 1 pad to avoid LDS bank conflicts

// ---------------------------------------------------------------------------
// Kernel 1: MC estimate of the kernel integral, 32 observers (16 per strategy)
// ---------------------------------------------------------------------------
__global__ void mc_integral_kernel(const float* __restrict__ mc,
                                   const float* __restrict__ c0, const float* __restrict__ s0,
                                   const float* __restrict__ c1, const float* __restrict__ s1,
                                   int M, float* __restrict__ integ) {
  int o = threadIdx.x;                       // 0..31
  const float* C = (o < OBS) ? c0 : c1;
  const float* S = (o < OBS) ? s0 : s1;
  int oo = o & (OBS - 1);
  float cx = C[oo * 3 + 0], cy = C[oo * 3 + 1], cz = C[oo * 3 + 2];
  float sg = S[oo];
  float inv = 1.0f / (2.0f * sg * sg);
  float acc = 0.0f;
  for (int m = 0; m < M; ++m) {
    float dx = mc[m * 3 + 0] * KSIZE - cx;
    float dy = mc[m * 3 + 1] * KSIZE - cy;
    float dz = mc[m * 3 + 2] * KSIZE - cz;
    acc += __expf(-(dx * dx + dy * dy + dz * dz) * inv);
  }
  integ[o] = acc / (float)M;
}

// ---------------------------------------------------------------------------
// Kernel 2: fused brute-force KNN (WMMA f32 16x16x4 distance tiles) +
//           Gaussian observer responses + 32->16 MLP + concat output.
// One wave owns 32 rows (points); lane L owns row r0+L and keeps a sorted
// top-16 (packed dist|idx keys) in registers.
// ---------------------------------------------------------------------------
__global__ __launch_bounds__(256)
void knn_gib_kernel(const float* __restrict__ coords,
                    const float* __restrict__ centers0, const float* __restrict__ sigmas0,
                    const float* __restrict__ centers1, const float* __restrict__ sigmas1,
                    const float* __restrict__ Wm, const float* __restrict__ bias,
                    const float* __restrict__ integ, float* __restrict__ out) {
  __shared__ float lds_tile[WAVES_PER_BLOCK][32 * TILE_STRIDE]; // per-wave 32x16 d2 tile
  __shared__ float lds_rowsq[WAVES_PER_BLOCK][32];

  const int lane  = threadIdx.x & 31;
  const int wave  = threadIdx.x >> 5;
  const int gwave = blockIdx.x * WAVES_PER_BLOCK + wave;
  const int wavesPerBatch = NPTS / 32;                 // 256
  const int b  = gwave / wavesPerBatch;
  const int r0 = (gwave % wavesPerBatch) * 32;
  const float* __restrict__ cb = coords + (size_t)b * NPTS * 3;

  const int hh = lane >> 4;                            // half of wave (0/1)
  const int ln = lane & 15;

  // ---- own row coords + squared norm (shared across wave via LDS) ----
  const int myrow = r0 + lane;
  float px = cb[myrow * 3 + 0];
  float py = cb[myrow * 3 + 1];
  float pz = cb[myrow * 3 + 2];
  float myq = px * px + py * py + pz * pz;
  lds_rowsq[wave][lane] = myq;
  asm volatile("s_wait_dscnt 0" ::: "memory");
  __builtin_amdgcn_wave_barrier();

  // ---- A tiles for WMMA: A = -2 * X_rows, padded K=4 ----
  // f32 16x4 A layout: lane<16 holds K=0(v0),K=1(v1); lane>=16 holds K=2(v0),K=3(v1)
  v2f aLo, aHi;
  {
    int rA = r0 + ln;
    float x = cb[rA * 3 + 0], y = cb[rA * 3 + 1], z = cb[rA * 3 + 2];
    aLo.x = hh ? (-2.0f * z) : (-2.0f * x);
    aLo.y = hh ? 0.0f        : (-2.0f * y);
    int rB = r0 + 16 + ln;
    float x2 = cb[rB * 3 + 0], y2 = cb[rB * 3 + 1], z2 = cb[rB * 3 + 2];
    aHi.x = hh ? (-2.0f * z2) : (-2.0f * x2);
    aHi.y = hh ? 0.0f         : (-2.0f * y2);
  }

  // row squared norms matching the C/D layout: M = v + 8*hh (+16 for hi tile)
  float rq0[8], rq1[8];
#pragma unroll
  for (int v = 0; v < 8; ++v) {
    rq0[v] = lds_rowsq[wave][v + 8 * hh];
    rq1[v] = lds_rowsq[wave][16 + v + 8 * hh];
  }

  // ---- top-16 sorted key list (ascending), keys = distbits|colidx ----
  unsigned heap[16];
#pragma unroll
  for (int i = 0; i < 16; ++i) heap[i] = 0xFFFFFFFFu;
  unsigned hmax = 0xFFFFFFFFu;

  float* tl = &lds_tile[wave][0];

  for (int t = 0; t < NPTS / 16; ++t) {
    const int col0 = t * 16;
    // B tile: B[k][n], k0=x,k1=y,k2=z,k3=0 ; lane half picks k-pair
    int cc = col0 + ln;
    float bx = cb[cc * 3 + 0], by = cb[cc * 3 + 1], bz = cb[cc * 3 + 2];
    v2f Bt;
    Bt.x = hh ? bz   : bx;
    Bt.y = hh ? 0.0f : by;
    // column squared norm assembled across the two wave halves
    float part  = hh ? (bz * bz) : (bx * bx + by * by);
    float other = __shfl_xor(part, 16, 32);
    float colsq = part + other;

    v8f c0v, c1v;
#pragma unroll
    for (int v = 0; v < 8; ++v) { c0v[v] = rq0[v] + colsq; c1v[v] = rq1[v] + colsq; }

    // d2 tiles: rows r0..r0+15 and r0+16..r0+31 vs cols col0..col0+15
    v8f d0 = __builtin_amdgcn_wmma_f32_16x16x4_f32(false, aLo, false, Bt,
                                                   (short)0, c0v, false, false);
    v8f d1 = __builtin_amdgcn_wmma_f32_16x16x4_f32(false, aHi, false, Bt,
                                                   (short)0, c1v, false, false);

    // transpose through LDS: lane holds col ln, rows v+8*hh (+16)
    asm volatile("" ::: "memory");
#pragma unroll
    for (int v = 0; v < 8; ++v) tl[(v + 8 * hh) * TILE_STRIDE + ln] = d0[v];
#pragma unroll
    for (int v = 0; v < 8; ++v) tl[(16 + v + 8 * hh) * TILE_STRIDE + ln] = d1[v];
    asm volatile("s_wait_dscnt 0" ::: "memory");
    __builtin_amdgcn_wave_barrier();

    // lane L consumes row L: 16 candidates
#pragma unroll
    for (int j = 0; j < 16; ++j) {
      float d = tl[lane * TILE_STRIDE + j];
      unsigned key = (__float_as_uint(d) & 0xFFFFE000u) | (unsigned)(col0 + j);
      bool take = key < hmax;
      if (__any((int)take)) {
        unsigned k2 = take ? key : 0xFFFFFFFFu;
#pragma unroll
        for (int i = 0; i < 16; ++i) {
          unsigned mn = (k2 < heap[i]) ? k2 : heap[i];
          unsigned mx = (k2 < heap[i]) ? heap[i] : k2;
          heap[i] = mn;
          k2 = mx;
        }
        hmax = heap[15];
      }
    }
    asm volatile("s_wait_dscnt 0" ::: "memory");
    __builtin_amdgcn_wave_barrier();
  }

  // ---- gather the 16 nearest neighbors' relative coords ----
  float relx[16], rely[16], relz[16];
#pragma unroll
  for (int j = 0; j < 16; ++j) {
    int idx = (int)(heap[j] & 0x1FFFu);
    relx[j] = cb[idx * 3 + 0] - px;
    rely[j] = cb[idx * 3 + 1] - py;
    relz[j] = cb[idx * 3 + 2] - pz;
  }

  // ---- Gaussian observer responses (strategy0: k=8, strategy1: k=16) ----
  float feat[32];
#pragma unroll
  for (int o = 0; o < OBS; ++o) {
    float c0x = centers0[o * 3 + 0], c0y = centers0[o * 3 + 1], c0z = centers0[o * 3 + 2];
    float c1x = centers1[o * 3 + 0], c1y = centers1[o * 3 + 1], c1z = centers1[o * 3 + 2];
    float sg0 = sigmas0[o], sg1 = sigmas1[o];
    float inv0 = 1.0f / (2.0f * sg0 * sg0);
    float inv1 = 1.0f / (2.0f * sg1 * sg1);
    float s0 = 0.0f, s1 = 0.0f;
#pragma unroll
    for (int j = 0; j < 16; ++j) {
      float dx1 = relx[j] - c1x, dy1 = rely[j] - c1y, dz1 = relz[j] - c1z;
      s1 += __expf(-(dx1 * dx1 + dy1 * dy1 + dz1 * dz1) * inv1);
      if (j < 8) {
        float dx0 = relx[j] - c0x, dy0 = rely[j] - c0y, dz0 = relz[j] - c0z;
        s0 += __expf(-(dx0 * dx0 + dy0 * dy0 + dz0 * dz0) * inv0);
      }
    }
    feat[o]       = (s0 * (1.0f / 8.0f)  - integ[o]);          // * (2-0)/2 = 1
    feat[16 + o]  = (s1 * (1.0f / 16.0f) - integ[16 + o]) * 0.5f; // * (2-1)/2
  }

  // ---- MLP: relu(feat @ W^T + b), then concat [x(16), feat(32)] ----
  float* op = out + ((size_t)b * NPTS + myrow) * 48;
#pragma unroll
  for (int o = 0; o < 16; ++o) {
    float acc = bias[o];
#pragma unroll
    for (int c = 0; c < 32; ++c) acc += Wm[o * 32 + c] * feat[c];
    op[o] = fmaxf(acc, 0.0f);
  }
#pragma unroll
  for (int c = 0; c < 32; ++c) op[16 + c] = feat[c];
}

// ---------------------------------------------------------------------------
extern "C" void kernel_launch(void* const* d_in, const int* in_sizes, int n_in,
                              void* d_out, int out_size, void* d_ws, size_t ws_size,
                              hipStream_t stream) {
  const float* coords  = (const float*)d_in[0];
  // d_in[1] = feats : unused by the reference computation
  const float* mc      = (const float*)d_in[2];
  const float* c0      = (const float*)d_in[3];
  const float* s0      = (const float*)d_in[4];
  const float* c1      = (const float*)d_in[5];
  const float* s1      = (const float*)d_in[6];
  const float* Wm      = (const float*)d_in[7];
  const float* bias    = (const float*)d_in[8];
  float* integ = (float*)d_ws;                 // 32 floats of scratch
  float* outp  = (float*)d_out;

  int M = in_sizes[2] / 3;

  mc_integral_kernel<<<1, 32, 0, stream>>>(mc, c0, s0, c1, s1, M, integ);

  const int total_waves = NBATCH * NPTS / 32;             // 1024
  const int blocks = total_waves / WAVES_PER_BLOCK;       // 128
  knn_gib_kernel<<<blocks, WAVES_PER_BLOCK * 32, 0, stream>>>(
      coords, c0, s0, c1, s1, Wm, bias, integ, outp);
}